// RelationalPolicyHead_36953898615233
// MI455X (gfx1250) — compile-verified
//
#include <hip/hip_runtime.h>
#include <math.h>
#include <stdint.h>

typedef float v2f __attribute__((ext_vector_type(2)));
typedef float v4f __attribute__((ext_vector_type(4)));
typedef float v8f __attribute__((ext_vector_type(8)));

#define HIDDEN  128
#define FEAT    256            // 2*HIDDEN
#define KSTEPS  64             // FEAT / 4  (K=4 per WMMA)
#define NTILES  8              // HIDDEN / 16
#define THREADS 256
#define WAVES   8              // THREADS / 32
#define RED_BLOCKS 256

// Dynamic LDS layout (bytes):
//   [0, 131072)                  : W1 packed as v2f pairs: w1p[p*128 + j] = (W1[2p][j], W1[2p+1][j])
//   [131072, 131072+wave*16384)  : per-wave staging, 16 rows x 256 floats (concat src||tgt)
__global__ void rph_logits_kernel(const float* __restrict__ emb,
                                  const int*   __restrict__ moves,   // [2][E]
                                  const float* __restrict__ W1,      // [256][128]
                                  const float* __restrict__ b1,      // [128]
                                  const float* __restrict__ W2,      // [128]
                                  const float* __restrict__ b2,      // [1]
                                  float*       __restrict__ logits,  // [E]
                                  int E)
{
    extern __shared__ float lds[];
    const int tid    = threadIdx.x;
    const int lane   = tid & 31;
    const int wave   = tid >> 5;
    const int laneLo = lane & 15;      // N (or M) index within tile
    const int laneHi = lane >> 4;      // selects K+2 half / M+8 half

    v2f*   w1p  = (v2f*)lds;                                   // 128x128 v2f = 128KB
    float* feat = lds + 2 * 128 * 128 + wave * (16 * FEAT);    // 16KB per wave

    // ---- cooperatively pack W1 into LDS in WMMA-B layout ----
    for (int i = tid; i < 128 * 128; i += THREADS) {
        const int p = i >> 7;          // K-pair index (k = 2p, 2p+1)
        const int j = i & 127;         // N column
        v2f v;
        v.x = W1[(2 * p)     * HIDDEN + j];
        v.y = W1[(2 * p + 1) * HIDDEN + j];
        w1p[i] = v;
    }
    __syncthreads();

    // per-lane bias / W2 slices: column = 16*n + laneLo
    float b1r[NTILES], w2r[NTILES];
#pragma unroll
    for (int n = 0; n < NTILES; ++n) {
        b1r[n] = b1[n * 16 + laneLo];
        w2r[n] = W2[n * 16 + laneLo];
    }
    const float bias2 = b2[0];

    const int nTiles = E >> 4;         // E is a multiple of 16
    const int stride = gridDim.x * WAVES;
    const v2f* featv = (const v2f*)feat;   // row stride = FEAT/2 v2f

    // async-gather 16 edges' concatenated rows into this wave's slab.
    // lane L moves 16B at column 4L of each 128-float half-row (32 lanes * 16B = 512B)
    auto issueGather = [&](int base) {
#pragma unroll 4
        for (int r = 0; r < 16; ++r) {
            const int s = moves[base + r];       // source node id
            const int t = moves[E + base + r];   // target node id
            const float* gs = emb + (size_t)s * HIDDEN + 4 * lane;
            const float* gt = emb + (size_t)t * HIDDEN + 4 * lane;
            const uint32_t ds_s = (uint32_t)(uintptr_t)(feat + r * FEAT + 4 * lane);
            const uint32_t ds_t = (uint32_t)(uintptr_t)(feat + r * FEAT + HIDDEN + 4 * lane);
            asm volatile("global_load_async_to_lds_b128 %0, %1, off"
                         :: "v"(ds_s), "v"(gs) : "memory");
            asm volatile("global_load_async_to_lds_b128 %0, %1, off"
                         :: "v"(ds_t), "v"(gt) : "memory");
        }
    };

    // stage helpers: load A (1x v2f) + 8x B (v2f) for one K-step
#define LOAD_STAGE(KS, AREG, BARR)                                         \
    do {                                                                   \
        AREG = featv[laneLo * (FEAT / 2) + 2 * (KS) + laneHi];             \
        const v2f* bb_ = w1p + (2 * (KS) + laneHi) * HIDDEN + laneLo;      \
        _Pragma("unroll")                                                  \
        for (int n_ = 0; n_ < NTILES; ++n_) BARR[n_] = bb_[n_ * 16];       \
    } while (0)

#define MM_STAGE(AREG, BARR)                                               \
    do {                                                                   \
        _Pragma("unroll")                                                  \
        for (int n_ = 0; n_ < NTILES; ++n_)                                \
            c[n_] = __builtin_amdgcn_wmma_f32_16x16x4_f32(                 \
                false, AREG, false, BARR[n_], (short)0, c[n_],             \
                false, false);                                             \
    } while (0)

    int tile = blockIdx.x * WAVES + wave;
    if (tile < nTiles) issueGather(tile * 16);   // prime the pipeline

    for (; tile < nTiles; tile += stride) {
        const int base = tile * 16;
        asm volatile("s_wait_asynccnt 0" ::: "memory");   // slab ready

        // ---- 16x128 = A(16x256) * W1(256x128) via f32 WMMA, 2-stage pipeline ----
        v8f c[NTILES] = {};
        v2f aCur, aNxt;
        v2f bCur[NTILES], bNxt[NTILES];

        LOAD_STAGE(0, aCur, bCur);
        for (int ks = 0; ks < KSTEPS; ks += 2) {
            LOAD_STAGE(ks + 1, aNxt, bNxt);      // prefetch odd step
            MM_STAGE(aCur, bCur);                // compute even step
            if (ks + 2 < KSTEPS)
                LOAD_STAGE(ks + 2, aCur, bCur);  // prefetch next even step
            MM_STAGE(aNxt, bNxt);                // compute odd step
        }

        // all slab ds-reads must have RETURNED before the async engine may
        // overwrite the slab (ASYNCcnt and DScnt are independent counters)
        asm volatile("s_wait_dscnt 0" ::: "memory");
        const int nxt = tile + stride;
        if (nxt < nTiles) issueGather(nxt * 16); // overlap gather with epilogue

        // ---- epilogue: bias + ReLU + dot with W2, per-edge logit ----
        // C layout: vgpr r, lane L -> element (M = r + 8*laneHi, N = 16n + laneLo)
        float acc[8];
#pragma unroll
        for (int r = 0; r < 8; ++r) acc[r] = 0.0f;
#pragma unroll
        for (int n = 0; n < NTILES; ++n) {
#pragma unroll
            for (int r = 0; r < 8; ++r) {
                float h = c[n][r] + b1r[n];
                h = h > 0.0f ? h : 0.0f;
                acc[r] += h * w2r[n];
            }
        }
        // reduce across the 16 lanes of each half (xor masks stay within halves)
#pragma unroll
        for (int r = 0; r < 8; ++r) {
            float a = acc[r];
            a += __shfl_xor(a, 1, 32);
            a += __shfl_xor(a, 2, 32);
            a += __shfl_xor(a, 4, 32);
            a += __shfl_xor(a, 8, 32);
            acc[r] = a;
        }
        if (laneLo == 0) {               // lanes 0 (edges 0..7) and 16 (edges 8..15)
            v4f lo = { acc[0] + bias2, acc[1] + bias2, acc[2] + bias2, acc[3] + bias2 };
            v4f hi = { acc[4] + bias2, acc[5] + bias2, acc[6] + bias2, acc[7] + bias2 };
            *(v4f*)(logits + base + laneHi * 8)     = lo;
            *(v4f*)(logits + base + laneHi * 8 + 4) = hi;
        }
    }
#undef LOAD_STAGE
#undef MM_STAGE
}

// ---------------- softmax (memory-bound, trivially small vs GEMM) ----------------

__global__ void rph_partial_max(const float* __restrict__ x, int n,
                                float* __restrict__ pmax) {
    __shared__ float sm[256];
    float m = -INFINITY;
    for (int i = blockIdx.x * blockDim.x + threadIdx.x; i < n;
         i += gridDim.x * blockDim.x)
        m = fmaxf(m, x[i]);
    sm[threadIdx.x] = m;
    __syncthreads();
    for (int s = 128; s > 0; s >>= 1) {
        if (threadIdx.x < s) sm[threadIdx.x] = fmaxf(sm[threadIdx.x], sm[threadIdx.x + s]);
        __syncthreads();
    }
    if (threadIdx.x == 0) pmax[blockIdx.x] = sm[0];
}

__global__ void rph_final_max(const float* __restrict__ pmax, int n,
                              float* __restrict__ gmax) {
    __shared__ float sm[256];
    float m = (threadIdx.x < n) ? pmax[threadIdx.x] : -INFINITY;
    sm[threadIdx.x] = m;
    __syncthreads();
    for (int s = 128; s > 0; s >>= 1) {
        if (threadIdx.x < s) sm[threadIdx.x] = fmaxf(sm[threadIdx.x], sm[threadIdx.x + s]);
        __syncthreads();
    }
    if (threadIdx.x == 0) gmax[0] = sm[0];
}

__global__ void rph_exp_psum(const float* __restrict__ logits,
                             const float* __restrict__ gmax, int n,
                             float* __restrict__ out, float* __restrict__ psum) {
    __shared__ float sm[256];
    const float m = gmax[0];
    float s = 0.0f;
    for (int i = blockIdx.x * blockDim.x + threadIdx.x; i < n;
         i += gridDim.x * blockDim.x) {
        float e = __expf(logits[i] - m);
        out[i] = e;
        s += e;
    }
    sm[threadIdx.x] = s;
    __syncthreads();
    for (int t = 128; t > 0; t >>= 1) {
        if (threadIdx.x < t) sm[threadIdx.x] += sm[threadIdx.x + t];
        __syncthreads();
    }
    if (threadIdx.x == 0) psum[blockIdx.x] = sm[0];
}

__global__ void rph_final_sum(const float* __restrict__ psum, int n,
                              float* __restrict__ ginv) {
    __shared__ float sm[256];
    sm[threadIdx.x] = (threadIdx.x < n) ? psum[threadIdx.x] : 0.0f;
    __syncthreads();
    for (int t = 128; t > 0; t >>= 1) {
        if (threadIdx.x < t) sm[threadIdx.x] += sm[threadIdx.x + t];
        __syncthreads();
    }
    if (threadIdx.x == 0) ginv[0] = 1.0f / sm[0];
}

__global__ void rph_scale(float* __restrict__ out,
                          const float* __restrict__ ginv, int n) {
    const float inv = ginv[0];
    for (int i = blockIdx.x * blockDim.x + threadIdx.x; i < n;
         i += gridDim.x * blockDim.x)
        out[i] *= inv;
}

extern "C" void kernel_launch(void* const* d_in, const int* in_sizes, int n_in,
                              void* d_out, int out_size, void* d_ws, size_t ws_size,
                              hipStream_t stream) {
    const float* emb   = (const float*)d_in[0];
    const int*   moves = (const int*)  d_in[1];
    const float* W1    = (const float*)d_in[2];
    const float* b1    = (const float*)d_in[3];
    const float* W2    = (const float*)d_in[4];
    const float* b2    = (const float*)d_in[5];
    float*       out   = (float*)d_out;

    const int E = in_sizes[1] / 2;

    // workspace: logits[E] | pmax[256] | psum[256] | gmax[1] | ginv[1]
    float* logits = (float*)d_ws;
    float* pmax   = logits + E;
    float* psum   = pmax + RED_BLOCKS;
    float* gmax   = psum + RED_BLOCKS;
    float* ginv   = gmax + 1;

    const int nTiles = E >> 4;
    int blocks = (nTiles + WAVES - 1) / WAVES;
    if (blocks > 512) blocks = 512;
    if (blocks < 1) blocks = 1;

    const size_t ldsBytes = (size_t)(2 * 128 * 128 + WAVES * 16 * FEAT) * sizeof(float); // 256KB

    rph_logits_kernel<<<blocks, THREADS, ldsBytes, stream>>>(emb, moves, W1, b1,
                                                             W2, b2, logits, E);
    rph_partial_max<<<RED_BLOCKS, 256, 0, stream>>>(logits, E, pmax);
    rph_final_max<<<1, 256, 0, stream>>>(pmax, RED_BLOCKS, gmax);
    rph_exp_psum<<<RED_BLOCKS, 256, 0, stream>>>(logits, gmax, E, out, psum);
    rph_final_sum<<<1, 256, 0, stream>>>(psum, RED_BLOCKS, ginv);
    rph_scale<<<(E + 255) / 256, 256, 0, stream>>>(out, ginv, E);
}